// Spline_42932493091552
// MI455X (gfx1250) — compile-verified
//
#include <hip/hip_runtime.h>
#include <stdint.h>

// ---------------- problem constants (from reference) ----------------
#define B_ROWS   65536
#define DIM      256
#define NBINS    32
#define MINW     1e-3f      // MIN_BIN_WIDTH == MIN_BIN_HEIGHT
#define MIND     1e-3f      // MIN_DERIVATIVE

// ---------------- d_ws table layout (floats) ----------------
//   cumw rows  [0..32]  : cumwidths[k][DIM]
//   deriv rows [33..65] : derivatives[k][DIM]
//   quad (float4)       : per (bin, dim): (cumheight, height, delta, deriv0)
#define CUMW_OFF   0
#define DERIV_OFF  (33 * DIM)
#define QUAD_OFF   (66 * DIM)            // float offset; float4 array starts here
// total = 66*256 + 32*256*4 = 49,664 floats = 198,656 bytes

#define COLS      32                      // dims per apply-block
#define RPI       8                       // rows per iteration (256 threads / 32 cols)
#define GRIDY     256
#define ROW_ITERS (B_ROWS / (GRIDY * RPI))   // 32

#ifndef __has_builtin
#define __has_builtin(x) 0
#endif

#if __has_builtin(__builtin_amdgcn_global_load_async_to_lds_b128)
#define HAVE_ASYNC_LDS 1
#else
#define HAVE_ASYNC_LDS 0
#endif

// async builtin operand types: (v4i addrspace(1)*, v4i addrspace(3)*, imm, imm)
typedef int v4i __attribute__((ext_vector_type(4)));
typedef __attribute__((address_space(1))) v4i* g_v4i_p;
typedef __attribute__((address_space(3))) v4i* l_v4i_p;

__device__ __forceinline__ void lds_copy16(const float* g, float* l) {
#if HAVE_ASYNC_LDS
    __builtin_amdgcn_global_load_async_to_lds_b128(
        (g_v4i_p)(uintptr_t)g, (l_v4i_p)(uint32_t)(uintptr_t)l, 0, 0);
#else
    *(float4*)l = *(const float4*)g;
#endif
}
__device__ __forceinline__ void lds_copy_fence() {
#if HAVE_ASYNC_LDS
#if __has_builtin(__builtin_amdgcn_s_wait_asynccnt)
    __builtin_amdgcn_s_wait_asynccnt(0);
#else
    asm volatile("s_wait_asynccnt 0" ::: "memory");
#endif
#endif
}

// ---------------- wave32 helpers ----------------
__device__ __forceinline__ float wave_max32(float v) {
#pragma unroll
    for (int s = 16; s >= 1; s >>= 1) v = fmaxf(v, __shfl_xor(v, s, 32));
    return v;
}
__device__ __forceinline__ float wave_sum32(float v) {
#pragma unroll
    for (int s = 16; s >= 1; s >>= 1) v += __shfl_xor(v, s, 32);
    return v;
}
__device__ __forceinline__ float softplusf(float x) {
    // jax.nn.softplus: max(x,0) + log1p(exp(-|x|))
    return fmaxf(x, 0.0f) + log1pf(expf(-fabsf(x)));
}

// =====================================================================
// Kernel 1: normalize spline params -> table in d_ws.
// One wave (32 lanes) per dim; lane j owns bin j. 256 blocks x 32 threads.
// =====================================================================
__global__ void __launch_bounds__(NBINS)
spline_precompute(const float* __restrict__ width,
                  const float* __restrict__ height,
                  const float* __restrict__ deriv,
                  float* __restrict__ tab) {
    const int d = blockIdx.x;    // dim
    const int j = threadIdx.x;   // bin

    // ---- widths: softmax -> floor -> inclusive scan -> clamp endpoints ----
    float uw  = width[d * NBINS + j];
    float ew  = expf(uw - wave_max32(uw));
    float szw = MINW + (1.0f - MINW * (float)NBINS) * (ew / wave_sum32(ew));
    float cw  = szw;
#pragma unroll
    for (int s = 1; s < 32; s <<= 1) { float v = __shfl_up(cw, s, 32); if (j >= s) cw += v; }
    float cw_next = (j == NBINS - 1) ? 1.0f : cw;          // cumwidths[j+1]
    float cw_prev = __shfl_up(cw_next, 1, 32);
    if (j == 0) cw_prev = 0.0f;                             // cumwidths[j]
    float wj = cw_next - cw_prev;                           // widths from diffs
    tab[CUMW_OFF + (j + 1) * DIM + d] = cw_next;
    if (j == 0) tab[CUMW_OFF + d] = 0.0f;

    // ---- heights ----
    float uh  = height[d * NBINS + j];
    float eh  = expf(uh - wave_max32(uh));
    float szh = MINW + (1.0f - MINW * (float)NBINS) * (eh / wave_sum32(eh));
    float chs = szh;
#pragma unroll
    for (int s = 1; s < 32; s <<= 1) { float v = __shfl_up(chs, s, 32); if (j >= s) chs += v; }
    float ch_next = (j == NBINS - 1) ? 1.0f : chs;
    float ch_prev = __shfl_up(ch_next, 1, 32);
    if (j == 0) ch_prev = 0.0f;                             // cumheights[j]
    float hj = ch_next - ch_prev;

    // ---- derivatives: pad with const so boundary derivative == 1 ----
    float cst = logf(expm1f(1.0f - MIND));
    float ud  = (j == 0) ? cst : deriv[d * (NBINS - 1) + (j - 1)];
    float dj  = MIND + softplusf(ud);
    tab[DERIV_OFF + j * DIM + d] = dj;
    if (j == NBINS - 1)
        tab[DERIV_OFF + NBINS * DIM + d] = MIND + softplusf(cst);

    // ---- packed per-bin quad: (cumheight_lo, height, delta, deriv_lo) ----
    float4* qtab = (float4*)(tab + QUAD_OFF);
    float4 q; q.x = ch_prev; q.y = hj; q.z = hj / wj; q.w = dj;
    qtab[j * DIM + d] = q;
}

// =====================================================================
// Kernel 2: streaming spline apply. grid = (DIM/32, 256), block = 256.
// blockIdx.x picks a 32-dim column group; 24.8 KB table slice is staged
// into LDS via async global->LDS b128 (ASYNCcnt), then rows stream.
// =====================================================================
__global__ void __launch_bounds__(256)
spline_apply(const float* __restrict__ x,
             const float* __restrict__ tab,
             float* __restrict__ y,
             float* __restrict__ logdet) {
    __shared__ __align__(16) float  cumwS[33 * COLS];    //  4,224 B
    __shared__ __align__(16) float  derivS[33 * COLS];   //  4,224 B
    __shared__ __align__(16) float4 quadS[NBINS * COLS]; // 16,384 B

    const int tid  = threadIdx.x;
    const int cg   = blockIdx.x;                  // column group (0..7)
    const int col  = tid & (COLS - 1);
    const int rof  = tid >> 5;                    // 0..7 rows per iteration
    const int gcol = cg * COLS + col;

    // -------- stage table slice into LDS (async b128) --------
    for (int i = tid; i < 33 * (COLS / 4); i += 256) {          // cumw rows
        int r = i >> 3, q = (i & 7) << 2;
        lds_copy16(tab + CUMW_OFF + r * DIM + cg * COLS + q, &cumwS[r * COLS + q]);
    }
    for (int i = tid; i < 33 * (COLS / 4); i += 256) {          // deriv rows
        int r = i >> 3, q = (i & 7) << 2;
        lds_copy16(tab + DERIV_OFF + r * DIM + cg * COLS + q, &derivS[r * COLS + q]);
    }
    const float* quadG = tab + QUAD_OFF;
    for (int i = tid; i < NBINS * COLS; i += 256) {             // quads (1 float4 each)
        int b = i >> 5, c = i & (COLS - 1);
        lds_copy16(quadG + 4 * (b * DIM + cg * COLS + c), (float*)&quadS[b * COLS + c]);
    }
    lds_copy_fence();
    __syncthreads();

    // -------- stream rows --------
#pragma unroll 1
    for (int it = 0; it < ROW_ITERS; ++it) {
        const int row = ((it * GRIDY + blockIdx.y) << 3) + rof;
        const int idx = row * DIM + gcol;

        float xv = __builtin_nontemporal_load(x + idx);
        if (it + 1 < ROW_ITERS)
            __builtin_prefetch(x + idx + GRIDY * RPI * DIM, 0, 0);

        const bool  inside = (xv >= 0.0f) && (xv <= 1.0f);
        const float xc     = fminf(fmaxf(xv, 0.0f), 1.0f);

        // branchless binary search over cumwidths[0..32]
        int lo = 0;
#pragma unroll
        for (int s = 16; s >= 1; s >>= 1) {
            float c = cumwS[(lo + s) * COLS + col];
            if (c <= xc) lo += s;
        }

        const float  c_lo = cumwS[lo * COLS + col];
        const float  c_hi = cumwS[(lo + 1) * COLS + col];
        const float4 q    = quadS[lo * COLS + col];     // (ch, h, delta, d0)
        const float  d1   = derivS[(lo + 1) * COLS + col];

        const float w     = c_hi - c_lo;                // == widths[lo] bitwise
        const float theta = (xc - c_lo) * __builtin_amdgcn_rcpf(w);
        const float omt   = 1.0f - theta;
        const float t1m   = theta * omt;
        const float th2   = theta * theta;

        const float num  = q.y * fmaf(q.z, th2, q.w * t1m);
        const float den  = fmaf(q.w + d1 - 2.0f * q.z, t1m, q.z);
        const float rden = __builtin_amdgcn_rcpf(den);
        const float yin  = q.x + num * rden;

        const float dn = (q.z * q.z) * fmaf(d1, th2, fmaf(2.0f * q.z, t1m, q.w * omt * omt));
        const float ld = __logf(dn * rden * rden);      // log(dn) - 2*log(den)

        __builtin_nontemporal_store(inside ? yin : xv, y + idx);
        __builtin_nontemporal_store(inside ? ld : 0.0f, logdet + idx);
    }
}

// =====================================================================
extern "C" void kernel_launch(void* const* d_in, const int* in_sizes, int n_in,
                              void* d_out, int out_size, void* d_ws, size_t ws_size,
                              hipStream_t stream) {
    (void)in_sizes; (void)n_in; (void)out_size; (void)ws_size;
    const float* x          = (const float*)d_in[0];   // [B, DIM]
    const float* width      = (const float*)d_in[1];   // [DIM, 32]
    const float* height     = (const float*)d_in[2];   // [DIM, 32]
    const float* derivative = (const float*)d_in[3];   // [DIM, 31]

    float* y      = (float*)d_out;                      // [B, DIM]
    float* logdet = y + (size_t)B_ROWS * DIM;           // [B, DIM]
    float* tab    = (float*)d_ws;                       // 49,664 floats

    spline_precompute<<<dim3(DIM), dim3(NBINS), 0, stream>>>(width, height, derivative, tab);

    dim3 grid(DIM / COLS, GRIDY);
    spline_apply<<<grid, dim3(256), 0, stream>>>(x, tab, y, logdet);
}